// TransMIL_60224031425185
// MI455X (gfx1250) — compile-verified
//
#include <hip/hip_runtime.h>

// ============================================================================
// TransMIL forward for MI455X (gfx1250, wave32).  All large GEMMs go through
// V_WMMA_F32_16X16X4_F32 (exact fp32 matrix op).  Wave = 32x32 output tile,
// block = 4 waves = 64x64 tile, batched via blockIdx.z.
// ============================================================================

typedef float v2f __attribute__((ext_vector_type(2)));
typedef float v8f __attribute__((ext_vector_type(8)));

#define CDIV(a,b) (((a)+(b)-1)/(b))

// ----------------------- problem constants ---------------------------------
constexpr int BB    = 2;
constexpr int NTOK  = 4096;
constexpr int NH    = 4097;    // tokens + cls
constexpr int NPAD  = 4352;    // padded to multiple of 256 landmarks
constexpr int PADF  = 255;     // front padding = NPAD - NH
constexpr int DIMC  = 512;
constexpr int NHEADS= 8;
constexpr int DHD   = 64;
constexpr int LM    = 256;     // landmarks
constexpr int LFAC  = 17;      // NPAD / LM

// ----------------------- workspace layout (floats) --------------------------
constexpr long long SZ_H    = 1LL*BB*NH*DIMC;          // 4,195,328
constexpr long long SZ_XPAD = 1LL*BB*NPAD*DIMC;        // 4,456,448
constexpr long long SZ_HB   = 1LL*BB*NHEADS*NPAD*DHD;  // 4,456,448
constexpr long long SZ_L    = 1LL*BB*NHEADS*LM*DHD;    //   262,144
constexpr long long SZ_A1   = 1LL*BB*NHEADS*NPAD*LM;   // 17,825,792
constexpr long long SZ_SQ   = 1LL*BB*NHEADS*LM*LM;     //  1,048,576

constexpr long long OF_H     = 0;
constexpr long long OF_XPAD  = OF_H + SZ_H;
constexpr long long OF_QKV   = OF_XPAD + SZ_XPAD;
constexpr long long OF_A1    = OF_XPAD;            // overlays xpad+qkv (same size)
constexpr long long OF_Q     = OF_XPAD + SZ_A1;
constexpr long long OF_K     = OF_Q + SZ_HB;
constexpr long long OF_V     = OF_K + SZ_HB;
constexpr long long OF_QL    = OF_V + SZ_HB;
constexpr long long OF_KL    = OF_QL + SZ_L;
constexpr long long OF_A3    = OF_KL + SZ_L;
constexpr long long OF_HLN   = OF_A3;              // LN output, dead before a3
constexpr long long OF_ATTN  = OF_A3;              // after a3 consumed
constexpr long long OF_MERGED= OF_A3 + SZ_HB;
constexpr long long OF_OUTP  = OF_A3 + 2*SZ_HB;
constexpr long long OF_SQ0   = OF_A3 + SZ_A1;      // a2,z,az,t,m (5 square bufs)
constexpr long long OF_A3V   = OF_SQ0 + 5*SZ_SQ;
constexpr long long OF_PT    = OF_A3V + SZ_L;
constexpr long long OF_SMALL = OF_PT + SZ_L;
constexpr long long OF_RS    = OF_SMALL;           // 4096 row sums
constexpr long long OF_CS    = OF_RS + 4096;       // 4096 col sums
constexpr long long OF_SCAL  = OF_CS + 4096;       // 2 scalars
constexpr long long OF_HN    = OF_SCAL + 16;       // (2,512) final cls features
constexpr long long OF_IDX   = OF_HN + 1024;       // 256*4 ints
// GCN overlays (OF_V region is dead after attention 2)
constexpr long long OF_XC    = OF_V;
constexpr long long OF_TDS   = OF_XC  + 256*512;
constexpr long long OF_XG    = OF_TDS + 256*256;
constexpr long long OF_XN    = OF_XG  + 256*512;
constexpr long long OF_S     = OF_XN  + 256*512;
constexpr long long OF_EDGE  = OF_S   + 256*256;
constexpr long long OF_G1    = OF_EDGE+ 256*512;
constexpr long long OF_G2    = OF_G1  + 256*512;
constexpr long long OF_T1B   = OF_G2  + 256*512;
constexpr long long OF_H1G   = OF_T1B + 256*512;
constexpr long long OF_GM    = OF_H1G + 256*512;
constexpr long long OF_T2B   = OF_GM  + 256*512;
constexpr long long OF_H2G   = OF_T2B + 256*512;
// PPEG overlays (OF_Q/OF_K are dead between attentions)
constexpr long long OF_CNN1  = OF_Q;
constexpr long long OF_CNN2  = OF_K;

// ============================================================================
// WMMA GEMM:  C = act(alpha * A(MxK,row) * op(B) + bias)
//   BT=true : B stored N x K row-major (computes A * B^T, the `x @ W.T` case)
//   BT=false: B stored K x N row-major (plain A * B)
// Requires M%64==0, N%64==0, K%16==0. Batched via blockIdx.z with strides.
// ============================================================================
__device__ __forceinline__ void wmma_store(float* __restrict__ C, v8f acc,
                                           int m0, int n0, int N, int hf, int l16,
                                           const float* __restrict__ bias,
                                           float alpha, int ACT)
{
  const int nb = n0 + l16;
  const float bv = bias ? bias[nb] : 0.f;
#pragma unroll
  for (int vv = 0; vv < 8; ++vv) {
    const int mr = m0 + vv + 8*hf;       // C layout: lanes 0-15 -> M=vgpr, 16-31 -> M=vgpr+8
    float val = alpha * acc[vv] + bv;
    if (ACT == 1) val = fmaxf(val, 0.f);
    if (ACT == 2) val = (val >= 0.f) ? val : 0.01f * val;
    C[(long long)mr * N + nb] = val;
  }
}

template<bool BT, int ACT>
__launch_bounds__(128)
__global__ void k_gemm(const float* __restrict__ A, const float* __restrict__ B,
                       const float* __restrict__ bias, float* __restrict__ C,
                       int N, int K, long long sA, long long sB, long long sC,
                       float alpha)
{
  const long long bz = blockIdx.z;
  A += bz * sA;  B += bz * sB;  C += bz * sC;
  const int wave = threadIdx.x >> 5;
  const int lane = threadIdx.x & 31;
  const int hf   = lane >> 4;          // half-wave selects K pair / M+8
  const int l16  = lane & 15;
  const int m0 = blockIdx.y * 64 + (wave >> 1) * 32;
  const int n0 = blockIdx.x * 64 + (wave & 1) * 32;

  v8f acc00 = {}, acc01 = {}, acc10 = {}, acc11 = {};
  const float* Ar0 = A + (long long)(m0 + l16) * K;
  const float* Ar1 = A + (long long)(m0 + 16 + l16) * K;

#pragma unroll 4
  for (int k = 0; k < K; k += 4) {
    const int kk = k + 2*hf;
    // A 16x4 fragment: lane holds row M=l16, cols (kk, kk+1)
    v2f a0 = *(const v2f*)(Ar0 + kk);
    v2f a1 = *(const v2f*)(Ar1 + kk);
    v2f b0, b1;
    if (BT) {   // B is NxK row-major -> B^T fragment is contiguous
      b0 = *(const v2f*)(B + (long long)(n0 + l16) * K + kk);
      b1 = *(const v2f*)(B + (long long)(n0 + 16 + l16) * K + kk);
    } else {    // B is KxN row-major
      const float* Br0 = B + (long long)kk * N;
      const float* Br1 = B + (long long)(kk + 1) * N;
      b0.x = Br0[n0 + l16];        b0.y = Br1[n0 + l16];
      b1.x = Br0[n0 + 16 + l16];   b1.y = Br1[n0 + 16 + l16];
    }
    acc00 = __builtin_amdgcn_wmma_f32_16x16x4_f32(false, a0, false, b0, (short)0, acc00, false, false);
    acc01 = __builtin_amdgcn_wmma_f32_16x16x4_f32(false, a0, false, b1, (short)0, acc01, false, false);
    acc10 = __builtin_amdgcn_wmma_f32_16x16x4_f32(false, a1, false, b0, (short)0, acc10, false, false);
    acc11 = __builtin_amdgcn_wmma_f32_16x16x4_f32(false, a1, false, b1, (short)0, acc11, false, false);
  }
  wmma_store(C, acc00, m0,      n0,      N, hf, l16, bias, alpha, ACT);
  wmma_store(C, acc01, m0,      n0 + 16, N, hf, l16, bias, alpha, ACT);
  wmma_store(C, acc10, m0 + 16, n0,      N, hf, l16, bias, alpha, ACT);
  wmma_store(C, acc11, m0 + 16, n0 + 16, N, hf, l16, bias, alpha, ACT);
}

static inline void gemm(hipStream_t st, bool bt, int act,
                        const float* A, const float* B, const float* bias, float* C,
                        int M, int N, int K,
                        long long sA, long long sB, long long sC, int batch,
                        float alpha = 1.f)
{
  dim3 g(N/64, M/64, batch), blk(128);
  if (bt) {
    if (act == 0)      k_gemm<true ,0><<<g,blk,0,st>>>(A,B,bias,C,N,K,sA,sB,sC,alpha);
    else if (act == 1) k_gemm<true ,1><<<g,blk,0,st>>>(A,B,bias,C,N,K,sA,sB,sC,alpha);
    else               k_gemm<true ,2><<<g,blk,0,st>>>(A,B,bias,C,N,K,sA,sB,sC,alpha);
  } else {
    k_gemm<false,0><<<g,blk,0,st>>>(A,B,bias,C,N,K,sA,sB,sC,alpha);
  }
}

// ============================================================================
// Elementwise / reduction kernels
// ============================================================================
__global__ void k_layernorm(const float* __restrict__ X, long long sx,
                            float* __restrict__ Y, long long sy,
                            const float* __restrict__ g, const float* __restrict__ b,
                            int C)
{
  const float* x = X + (long long)blockIdx.x * sx;
  float* y       = Y + (long long)blockIdx.x * sy;
  __shared__ float red[256];
  const int t = threadIdx.x;
  float s = 0.f;
  for (int i = t; i < C; i += 256) s += x[i];
  red[t] = s; __syncthreads();
  for (int st = 128; st > 0; st >>= 1) { if (t < st) red[t] += red[t+st]; __syncthreads(); }
  const float mu = red[0] / C;  __syncthreads();
  float v = 0.f;
  for (int i = t; i < C; i += 256) { float d = x[i] - mu; v += d*d; }
  red[t] = v; __syncthreads();
  for (int st = 128; st > 0; st >>= 1) { if (t < st) red[t] += red[t+st]; __syncthreads(); }
  const float inv = 1.f / sqrtf(red[0] / C + 1e-5f);
  for (int i = t; i < C; i += 256) y[i] = (x[i] - mu) * inv * g[i] + b[i];
}

__global__ void k_softmax(float* __restrict__ X, int L)
{
  float* x = X + (long long)blockIdx.x * L;
  __shared__ float red[256];
  const int t = threadIdx.x;
  float m = -3.402823466e38f;
  for (int i = t; i < L; i += 256) m = fmaxf(m, x[i]);
  red[t] = m; __syncthreads();
  for (int st = 128; st > 0; st >>= 1) { if (t < st) red[t] = fmaxf(red[t], red[t+st]); __syncthreads(); }
  m = red[0]; __syncthreads();
  float s = 0.f;
  for (int i = t; i < L; i += 256) { float e = expf(x[i] - m); x[i] = e; s += e; }
  red[t] = s; __syncthreads();
  for (int st = 128; st > 0; st >>= 1) { if (t < st) red[t] += red[t+st]; __syncthreads(); }
  const float inv = 1.f / red[0];
  for (int i = t; i < L; i += 256) x[i] *= inv;
}

__global__ void k_assembleh(const float* __restrict__ tmp, const float* __restrict__ cls,
                            float* __restrict__ h)
{
  long long t = blockIdx.x * 256LL + threadIdx.x;
  if (t >= SZ_H) return;
  const int c = (int)(t & (DIMC - 1));
  const long long r = t >> 9;
  const int n = (int)(r % NH);
  const int b = (int)(r / NH);
  h[t] = (n == 0) ? cls[c] : tmp[((long long)b*NTOK + (n-1))*DIMC + c];
}

__global__ void k_padfront(const float* __restrict__ src, float* __restrict__ dst)
{
  long long t = blockIdx.x * 256LL + threadIdx.x;
  if (t >= SZ_XPAD) return;
  const int c = (int)(t & (DIMC - 1));
  const long long r = t >> 9;
  const int n = (int)(r % NPAD);
  const int b = (int)(r / NPAD);
  dst[t] = (n < PADF) ? 0.f : src[((long long)b*NH + (n - PADF))*DIMC + c];
}

__global__ void k_split(const float* __restrict__ qkv, float* __restrict__ q,
                        float* __restrict__ k, float* __restrict__ v)
{
  long long t = blockIdx.x * 256LL + threadIdx.x;
  if (t >= SZ_HB) return;
  const int d = (int)(t & 63);
  const long long r = t >> 6;
  const int n = (int)(r % NPAD);
  const long long bh = r / NPAD;
  const int h = (int)(bh & 7);
  const int b = (int)(bh >> 3);
  const long long src = ((long long)b*NPAD + n)*(3*DIMC) + h*DHD + d;
  q[t] = qkv[src] * 0.125f;          // DH^-0.5
  k[t] = qkv[src + DIMC];
  v[t] = qkv[src + 2*DIMC];
}

__global__ void k_lmean(const float* __restrict__ x, float* __restrict__ xl)
{
  long long t = blockIdx.x * 256LL + threadIdx.x;
  if (t >= SZ_L) return;
  const int d = (int)(t & 63);
  const long long r = t >> 6;
  const int m = (int)(r % LM);
  const long long bh = r / LM;
  const float* base = x + (bh*NPAD + (long long)m*LFAC)*DHD + d;
  float s = 0.f;
#pragma unroll
  for (int j = 0; j < LFAC; ++j) s += base[j*DHD];
  xl[t] = s * (1.f / LFAC);
}

// ---- pinv helpers ----------------------------------------------------------
__global__ void k_pinvsums(const float* __restrict__ a, float* __restrict__ rs,
                           float* __restrict__ cs)
{
  int t = blockIdx.x * 256 + threadIdx.x;
  if (t >= BB*NHEADS*LM) return;
  const int i = t & (LM - 1);
  const long long bh = t >> 8;
  const float* ab = a + (bh << 16);
  float r = 0.f, c = 0.f;
  for (int j = 0; j < LM; ++j) { r += fabsf(ab[i*LM + j]); c += fabsf(ab[j*LM + i]); }
  rs[t] = r; cs[t] = c;
}

__global__ void k_redmax(const float* __restrict__ in, int n, float* __restrict__ out)
{
  __shared__ float red[256];
  const int t = threadIdx.x;
  float m = -3.402823466e38f;
  for (int i = t; i < n; i += 256) m = fmaxf(m, in[i]);
  red[t] = m; __syncthreads();
  for (int st = 128; st > 0; st >>= 1) { if (t < st) red[t] = fmaxf(red[t], red[t+st]); __syncthreads(); }
  if (t == 0) out[0] = red[0];
}

__global__ void k_pinvinit(const float* __restrict__ a, float* __restrict__ z,
                           const float* __restrict__ scal)
{
  long long t = blockIdx.x * 256LL + threadIdx.x;
  if (t >= SZ_SQ) return;
  const int j = (int)(t & (LM - 1));
  const int i = (int)((t >> 8) & (LM - 1));
  const long long bh = t >> 16;
  z[t] = a[(bh << 16) + (long long)j*LM + i] / (scal[0] * scal[1]);
}

__global__ void k_diaglin(const float* __restrict__ m, float* __restrict__ o, float c)
{
  long long t = blockIdx.x * 256LL + threadIdx.x;
  if (t >= SZ_SQ) return;
  const int j = (int)(t & (LM - 1));
  const int i = (int)((t >> 8) & (LM - 1));
  o[t] = ((i == j) ? c : 0.f) - m[t];
}

// ---- residual head conv (33-tap depthwise along sequence) ------------------
__global__ void k_resconv(float* __restrict__ attn, const float* __restrict__ v,
                          const float* __restrict__ rk)
{
  long long t = blockIdx.x * 256LL + threadIdx.x;
  if (t >= SZ_HB) return;
  const int d = (int)(t & 63);
  const long long r = t >> 6;
  const int n = (int)(r % NPAD);
  const long long bh = r / NPAD;
  const int h = (int)(bh & 7);
  const float* vb = v + bh * ((long long)NPAD * DHD);
  float s = 0.f;
#pragma unroll
  for (int tap = 0; tap < 33; ++tap) {
    const int nn = n + tap - 16;
    if (nn >= 0 && nn < NPAD) s += rk[h*33 + tap] * vb[(long long)nn*DHD + d];
  }
  attn[t] += s;
}

__global__ void k_merge(const float* __restrict__ attn, float* __restrict__ mrg)
{
  long long t = blockIdx.x * 256LL + threadIdx.x;
  if (t >= SZ_XPAD) return;
  const int c = (int)(t & (DIMC - 1));
  const long long r = t >> 9;
  const int n = (int)(r % NPAD);
  const int b = (int)(r / NPAD);
  const int h = c >> 6, d = c & 63;
  mrg[t] = attn[(((long long)(b*NHEADS + h))*NPAD + n)*DHD + d];
}

__global__ void k_resadd(float* __restrict__ h, const float* __restrict__ outp)
{
  long long t = blockIdx.x * 256LL + threadIdx.x;
  if (t >= SZ_H) return;
  const int c = (int)(t & (DIMC - 1));
  const long long r = t >> 9;
  const int n = (int)(r % NH);
  const int b = (int)(r / NH);
  h[t] += outp[((long long)b*NPAD + (PADF + n))*DIMC + c];   // out[:, -n:]
}

// ---- PPEG ------------------------------------------------------------------
__global__ void k_tocnn(const float* __restrict__ h, float* __restrict__ cnn)
{
  long long t = blockIdx.x * 256LL + threadIdx.x;
  if (t >= 1LL*BB*DIMC*4096) return;
  const int sp = (int)(t & 4095);
  const int c  = (int)((t >> 12) & (DIMC - 1));
  const int b  = (int)(t >> 21);
  cnn[t] = h[((long long)b*NH + 1 + sp)*DIMC + c];
}

__global__ void k_ppeg(const float* __restrict__ x, float* __restrict__ y,
                       const float* __restrict__ w7, const float* __restrict__ b7,
                       const float* __restrict__ w5, const float* __restrict__ b5,
                       const float* __restrict__ w3, const float* __restrict__ b3)
{
  long long t = blockIdx.x * 256LL + threadIdx.x;
  if (t >= 1LL*BB*DIMC*4096) return;
  const int xx = (int)(t & 63);
  const int yy = (int)((t >> 6) & 63);
  const int c  = (int)((t >> 12) & (DIMC - 1));
  const int b  = (int)(t >> 21);
  const float* xc = x + (((long long)b*DIMC + c) << 12);
  float acc = xc[(yy << 6) + xx] + b7[c] + b5[c] + b3[c];
#pragma unroll
  for (int p = 0; p < 7; ++p) { int iy = yy + p - 3; if (iy < 0 || iy >= 64) continue;
#pragma unroll
    for (int q = 0; q < 7; ++q) { int ix = xx + q - 3; if (ix < 0 || ix >= 64) continue;
      acc += xc[(iy << 6) + ix] * w7[c*49 + p*7 + q]; } }
#pragma unroll
  for (int p = 0; p < 5; ++p) { int iy = yy + p - 2; if (iy < 0 || iy >= 64) continue;
#pragma unroll
    for (int q = 0; q < 5; ++q) { int ix = xx + q - 2; if (ix < 0 || ix >= 64) continue;
      acc += xc[(iy << 6) + ix] * w5[c*25 + p*5 + q]; } }
#pragma unroll
  for (int p = 0; p < 3; ++p) { int iy = yy + p - 1; if (iy < 0 || iy >= 64) continue;
#pragma unroll
    for (int q = 0; q < 3; ++q) { int ix = xx + q - 1; if (ix < 0 || ix >= 64) continue;
      acc += xc[(iy << 6) + ix] * w3[c*9 + p*3 + q]; } }
  y[t] = acc;
}

__global__ void k_fromcnn(const float* __restrict__ cnn, float* __restrict__ h)
{
  long long t = blockIdx.x * 256LL + threadIdx.x;
  if (t >= 1LL*BB*DIMC*4096) return;
  const int sp = (int)(t & 4095);
  const int c  = (int)((t >> 12) & (DIMC - 1));
  const int b  = (int)(t >> 21);
  h[((long long)b*NH + 1 + sp)*DIMC + c] = cnn[t];
}

// ---- GCN branch ------------------------------------------------------------
__global__ void k_buildxc(const float* __restrict__ hn, const float* __restrict__ reh,
                          float* __restrict__ xc)
{
  int t = blockIdx.x * 256 + threadIdx.x;
  if (t >= 256*512) return;
  const int d = t & 511, i = t >> 9;
  xc[t] = (i < BB) ? hn[i*512 + d] : reh[(i - BB)*512 + d];
}

__global__ void k_rownorm(const float* __restrict__ X, float* __restrict__ Y, int C)
{
  const float* x = X + (long long)blockIdx.x * C;
  float* y       = Y + (long long)blockIdx.x * C;
  __shared__ float red[256];
  const int t = threadIdx.x;
  float s = 0.f;
  for (int i = t; i < C; i += 256) s += x[i] * x[i];
  red[t] = s; __syncthreads();
  for (int st = 128; st > 0; st >>= 1) { if (t < st) red[t] += red[t+st]; __syncthreads(); }
  const float inv = 1.f / sqrtf(red[0]);
  for (int i = t; i < C; i += 256) y[i] = x[i] * inv;
}

__global__ void k_topk4(const float* __restrict__ S, int* __restrict__ idx)
{
  int i = blockIdx.x * 256 + threadIdx.x;
  if (i >= 256) return;
  const float* row = S + i*256;
  int sel[4];
  for (int k = 0; k < 4; ++k) {
    float best = -3.402823466e38f; int bj = 0;
    for (int j = 0; j < 256; ++j) {
      bool used = false;
      for (int p = 0; p < k; ++p) used |= (sel[p] == j);
      if (!used && row[j] > best) { best = row[j]; bj = j; }
    }
    sel[k] = bj;
    idx[i*4 + k] = bj;
  }
}

__global__ void k_gathermean(const float* __restrict__ X, const int* __restrict__ idx,
                             float* __restrict__ out, int D)
{
  int t = blockIdx.x * 256 + threadIdx.x;
  if (t >= 256*D) return;
  const int d = t % D, i = t / D;
  float s = 0.f;
#pragma unroll
  for (int p = 0; p < 4; ++p) s += X[(long long)idx[i*4 + p]*D + d];
  out[t] = s * 0.25f;
}

__global__ void k_vadd(const float* __restrict__ a, const float* __restrict__ b,
                       float* __restrict__ o, int n)
{
  int t = blockIdx.x * 256 + threadIdx.x;
  if (t < n) o[t] = a[t] + b[t];
}

__global__ void k_addrelu(const float* __restrict__ a, const float* __restrict__ b,
                          float* __restrict__ o, int n)
{
  int t = blockIdx.x * 256 + threadIdx.x;
  if (t < n) o[t] = fmaxf(a[t] + b[t], 0.f);
}

__global__ void k_head(const float* __restrict__ X, const float* __restrict__ W,
                       const float* __restrict__ b, float* __restrict__ out,
                       int rows, int cols, int K)
{
  int t = blockIdx.x * blockDim.x + threadIdx.x;
  if (t >= rows*cols) return;
  const int i = t / cols, j = t % cols;
  float s = b[j];
  for (int kk = 0; kk < K; ++kk) s += X[(long long)i*K + kk] * W[(long long)j*K + kk];
  out[t] = s;
}

// ============================================================================
// Nystrom attention block (h += attn(LN(h)))
// ============================================================================
static void run_nystrom(hipStream_t st, float* ws,
                        const float* ln_g, const float* ln_b,
                        const float* qkv_w, const float* out_w,
                        const float* out_b, const float* res_k)
{
  float* h    = ws + OF_H;
  float* hln  = ws + OF_HLN;
  float* xpad = ws + OF_XPAD;
  float* qkv  = ws + OF_QKV;
  float* q    = ws + OF_Q;
  float* k    = ws + OF_K;
  float* v    = ws + OF_V;
  float* ql   = ws + OF_QL;
  float* kl   = ws + OF_KL;
  float* a1   = ws + OF_A1;
  float* a3   = ws + OF_A3;
  float* attn = ws + OF_ATTN;
  float* mrg  = ws + OF_MERGED;
  float* outp = ws + OF_OUTP;
  float* a2   = ws + OF_SQ0;
  float* z    = ws + OF_SQ0 + SZ_SQ;
  float* az   = ws + OF_SQ0 + 2*SZ_SQ;
  float* tb   = ws + OF_SQ0 + 3*SZ_SQ;
  float* mb   = ws + OF_SQ0 + 4*SZ_SQ;
  float* a3v  = ws + OF_A3V;
  float* pt   = ws + OF_PT;
  float* rs   = ws + OF_RS;
  float* cs   = ws + OF_CS;
  float* scal = ws + OF_SCAL;
  const long long ss = (long long)LM*LM;
  const int BH = BB*NHEADS;

  k_layernorm<<<BB*NH, 256, 0, st>>>(h, DIMC, hln, DIMC, ln_g, ln_b, DIMC);
  k_padfront<<<CDIV(SZ_XPAD,256), 256, 0, st>>>(hln, xpad);
  gemm(st, true, 0, xpad, qkv_w, nullptr, qkv, BB*NPAD, 3*DIMC, DIMC, 0,0,0, 1);
  k_split<<<CDIV(SZ_HB,256), 256, 0, st>>>(qkv, q, k, v);
  k_lmean<<<CDIV(SZ_L,256), 256, 0, st>>>(q, ql);
  k_lmean<<<CDIV(SZ_L,256), 256, 0, st>>>(k, kl);

  // a1 = softmax(q @ kl^T)   (bh, 4352, 256)  -- overlays xpad+qkv (now dead)
  gemm(st, true, 0, q, kl, nullptr, a1, NPAD, LM, DHD,
       (long long)NPAD*DHD, (long long)LM*DHD, (long long)NPAD*LM, BH);
  k_softmax<<<BH*NPAD, 256, 0, st>>>(a1, LM);
  // a2 = softmax(ql @ kl^T)  (bh, 256, 256)
  gemm(st, true, 0, ql, kl, nullptr, a2, LM, LM, DHD,
       (long long)LM*DHD, (long long)LM*DHD, ss, BH);
  k_softmax<<<BH*LM, 256, 0, st>>>(a2, LM);
  // a3 = softmax(ql @ k^T)   (bh, 256, 4352)
  gemm(st, true, 0, ql, k, nullptr, a3, LM, NPAD, DHD,
       (long long)LM*DHD, (long long)NPAD*DHD, (long long)LM*NPAD, BH);
  k_softmax<<<BH*LM, 256, 0, st>>>(a3, NPAD);

  // pinv(a2): z0 = a2^T / (max rowsum * max colsum); 6 Newton-Schulz iters
  k_pinvsums<<<CDIV(BH*LM,256), 256, 0, st>>>(a2, rs, cs);
  k_redmax<<<1, 256, 0, st>>>(rs, BH*LM, scal);
  k_redmax<<<1, 256, 0, st>>>(cs, BH*LM, scal + 1);
  k_pinvinit<<<CDIV(SZ_SQ,256), 256, 0, st>>>(a2, z, scal);
  for (int it = 0; it < 6; ++it) {
    gemm(st, false, 0, a2, z, nullptr, az, LM, LM, LM, ss, ss, ss, BH);
    k_diaglin<<<CDIV(SZ_SQ,256), 256, 0, st>>>(az, tb, 7.f);
    gemm(st, false, 0, az, tb, nullptr, mb, LM, LM, LM, ss, ss, ss, BH);
    k_diaglin<<<CDIV(SZ_SQ,256), 256, 0, st>>>(mb, tb, 15.f);
    gemm(st, false, 0, az, tb, nullptr, mb, LM, LM, LM, ss, ss, ss, BH);
    k_diaglin<<<CDIV(SZ_SQ,256), 256, 0, st>>>(mb, tb, 13.f);
    gemm(st, false, 0, z, tb, nullptr, az, LM, LM, LM, ss, ss, ss, BH, 0.25f);
    float* tsw = z; z = az; az = tsw;    // z now holds updated iterate
  }

  // a3v = a3 @ v ; pt = pinv @ a3v ; attn = a1 @ pt   (attn overlays dead a3)
  gemm(st, false, 0, a3, v, nullptr, a3v, LM, DHD, NPAD,
       (long long)LM*NPAD, (long long)NPAD*DHD, (long long)LM*DHD, BH);
  gemm(st, false, 0, z, a3v, nullptr, pt, LM, DHD, LM,
       ss, (long long)LM*DHD, (long long)LM*DHD, BH);
  gemm(st, false, 0, a1, pt, nullptr, attn, NPAD, DHD, LM,
       (long long)NPAD*LM, (long long)LM*DHD, (long long)NPAD*DHD, BH);
  k_resconv<<<CDIV(SZ_HB,256), 256, 0, st>>>(attn, v, res_k);
  k_merge<<<CDIV(SZ_XPAD,256), 256, 0, st>>>(attn, mrg);
  gemm(st, true, 0, mrg, out_w, out_b, outp, BB*NPAD, DIMC, DIMC, 0,0,0, 1);
  k_resadd<<<CDIV(SZ_H,256), 256, 0, st>>>(h, outp);
}

// ============================================================================
extern "C" void kernel_launch(void* const* d_in, const int* in_sizes, int n_in,
                              void* d_out, int out_size, void* d_ws, size_t ws_size,
                              hipStream_t stream)
{
  (void)in_sizes; (void)n_in; (void)out_size; (void)ws_size;
  const float* x       = (const float*)d_in[0];
  const float* fc1_w   = (const float*)d_in[1];
  const float* fc1_b   = (const float*)d_in[2];
  const float* cls_tok = (const float*)d_in[3];
  const float* ln1_g   = (const float*)d_in[4];
  const float* ln1_b   = (const float*)d_in[5];
  const float* qkv1_w  = (const float*)d_in[6];
  const float* out1_w  = (const float*)d_in[7];
  const float* out1_b  = (const float*)d_in[8];
  const float* res1_k  = (const float*)d_in[9];
  const float* ppeg_w7 = (const float*)d_in[10];
  const float* ppeg_b7 = (const float*)d_in[11];
  const float* ppeg_w5 = (const float*)d_in[12];
  const float* ppeg_b5 = (const float*)d_in[13];
  const float* ppeg_w3 = (const float*)d_in[14];
  const float* ppeg_b3 = (const float*)d_in[15];
  const float* ln2_g   = (const float*)d_in[16];
  const float* ln2_b   = (const float*)d_in[17];
  const float* qkv2_w  = (const float*)d_in[18];
  const float* out2_w  = (const float*)d_in[19];
  const float* out2_b  = (const float*)d_in[20];
  const float* res2_k  = (const float*)d_in[21];
  const float* norm_g  = (const float*)d_in[22];
  const float* norm_b  = (const float*)d_in[23];
  const float* fc2_w   = (const float*)d_in[24];
  const float* fc2_b   = (const float*)d_in[25];
  const float* dsl_w1  = (const float*)d_in[26];
  const float* dsl_b1  = (const float*)d_in[27];
  const float* dsl_w2  = (const float*)d_in[28];
  const float* dsl_b2  = (const float*)d_in[29];
  const float* gcn_w1  = (const float*)d_in[30];
  const float* gcn_we  = (const float*)d_in[31];
  const float* gcn_b1  = (const float*)d_in[32];
  const float* gcn_w2  = (const float*)d_in[33];
  const float* gcn_b2  = (const float*)d_in[34];
  const float* ghead_w = (const float*)d_in[35];
  const float* ghead_b = (const float*)d_in[36];
  const float* gdist_w = (const float*)d_in[37];
  const float* gdist_b = (const float*)d_in[38];
  const float* rehear  = (const float*)d_in[39];

  float* ws   = (float*)d_ws;
  float* fout = (float*)d_out;

  // ---- fc1 + relu, then prepend cls token  (Hs=64, add=0 for N=4096) ------
  float* tmp = ws + OF_XPAD;   // transient, region dead at this point
  gemm(stream, true, 1, x, fc1_w, fc1_b, tmp, BB*NTOK, DIMC, DIMC, 0,0,0, 1);
  k_assembleh<<<CDIV(SZ_H,256), 256, 0, stream>>>(tmp, cls_tok, ws + OF_H);

  // ---- attention 1 ---------------------------------------------------------
  run_nystrom(stream, ws, ln1_g, ln1_b, qkv1_w, out1_w, out1_b, res1_k);

  // ---- PPEG (depthwise 7/5/3 on 64x64 grid of features) -------------------
  const long long ncnn = 1LL*BB*DIMC*4096;
  k_tocnn  <<<CDIV(ncnn,256), 256, 0, stream>>>(ws + OF_H, ws + OF_CNN1);
  k_ppeg   <<<CDIV(ncnn,256), 256, 0, stream>>>(ws + OF_CNN1, ws + OF_CNN2,
                                                ppeg_w7, ppeg_b7, ppeg_w5, ppeg_b5,
                                                ppeg_w3, ppeg_b3);
  k_fromcnn<<<CDIV(ncnn,256), 256, 0, stream>>>(ws + OF_CNN2, ws + OF_H);

  // ---- attention 2 ---------------------------------------------------------
  run_nystrom(stream, ws, ln2_g, ln2_b, qkv2_w, out2_w, out2_b, res2_k);

  // ---- final norm on cls rows, mlp head ------------------------------------
  float* hn = ws + OF_HN;
  k_layernorm<<<BB, 256, 0, stream>>>(ws + OF_H, (long long)NH*DIMC, hn, DIMC,
                                      norm_g, norm_b, DIMC);
  k_head<<<1, 64, 0, stream>>>(hn, fc2_w, fc2_b, fout + 0, BB, 2, DIMC);

  // ---- graph branch --------------------------------------------------------
  float* xc  = ws + OF_XC;   float* tds = ws + OF_TDS;  float* xg  = ws + OF_XG;
  float* xn  = ws + OF_XN;   float* Sm  = ws + OF_S;    float* edge= ws + OF_EDGE;
  float* g1  = ws + OF_G1;   float* g2  = ws + OF_G2;   float* t1  = ws + OF_T1B;
  float* h1g = ws + OF_H1G;  float* gm  = ws + OF_GM;   float* t2  = ws + OF_T2B;
  float* h2g = ws + OF_H2G;
  int*   idx = (int*)(ws + OF_IDX);

  k_buildxc<<<CDIV(256*512,256), 256, 0, stream>>>(hn, rehear, xc);
  gemm(stream, true, 2, xc,  dsl_w1, dsl_b1, tds, 256, 256, 512, 0,0,0, 1);
  gemm(stream, true, 2, tds, dsl_w2, dsl_b2, xg,  256, 512, 256, 0,0,0, 1);
  k_rownorm<<<256, 256, 0, stream>>>(xg, xn, 512);
  gemm(stream, true, 0, xn, xn, nullptr, Sm, 256, 256, 512, 0,0,0, 1);
  k_topk4<<<1, 256, 0, stream>>>(Sm, idx);
  k_gathermean<<<CDIV(256*512,256), 256, 0, stream>>>(xg, idx, edge, 512);
  k_vadd<<<CDIV(256*512,256), 256, 0, stream>>>(xg, edge, t1, 256*512);
  gemm(stream, true, 0, t1,   gcn_w1, gcn_b1, g1, 256, 512, 512, 0,0,0, 1);
  gemm(stream, true, 0, edge, gcn_we, nullptr, g2, 256, 512, 512, 0,0,0, 1);
  k_addrelu<<<CDIV(256*512,256), 256, 0, stream>>>(g1, g2, h1g, 256*512);
  k_gathermean<<<CDIV(256*512,256), 256, 0, stream>>>(h1g, idx, gm, 512);
  k_vadd<<<CDIV(256*512,256), 256, 0, stream>>>(h1g, gm, t2, 256*512);
  gemm(stream, true, 0, t2, gcn_w2, gcn_b2, h2g, 256, 512, 512, 0,0,0, 1);
  k_head<<<1, 64, 0, stream>>>(h2g, ghead_w, ghead_b, fout + 4, BB, 2, DIMC);
  k_head<<<1, 64, 0, stream>>>(h1g, gdist_w, gdist_b, fout + 8, BB, 2, DIMC);
}